// RWEqualCELoss_50534585204961
// MI455X (gfx1250) — compile-verified
//
#include <hip/hip_runtime.h>

typedef __attribute__((ext_vector_type(2))) float v2f;
typedef __attribute__((ext_vector_type(8))) float v8f;

#define HBIG 1e10f
constexpr int BB = 4, CC = 4, HH = 256, WW = 256;
constexpr int HW = HH * WW;          // 65536
constexpr int NPIX = BB * HW;        // 262144 (B*H*W)
constexpr int NELEM = BB * CC * HW;  // 1048576
constexpr int NSLICE = BB * CC;      // 16

// Async global->LDS copy (gfx1250): one b32 per lane. The VDST operand is a
// VGPR holding the wave-relative LDS byte address; low 32 bits of a generic
// pointer to a __shared__ object are exactly that offset.
__device__ __forceinline__ void async_load_f32_to_lds(const float* gsrc, float* ldst) {
    unsigned loff = (unsigned)(size_t)ldst;
    unsigned long long ga = (unsigned long long)(size_t)gsrc;
    asm volatile("global_load_async_to_lds_b32 %0, %1, off"
                 :
                 : "v"(loff), "v"(ga)
                 : "memory");
}

// ---------------------------------------------------------------------------
// Kernel 1a: per-(b,c) slice, row pass of the EDT.
// mask = (y==c). f = BIG where mask else 0. g[i][j] = min_k f[i][k]+(j-k)^2
// Since f is binary, g is (distance to nearest zero in the row)^2, clamped to
// BIG when the row has no zero (matches reference exactly: BIG + 0).
// Output written TRANSPOSED: gT[bc][j][i] so the column pass reads rows.
// ---------------------------------------------------------------------------
__global__ void edt_rows(const int* __restrict__ y, float* __restrict__ gT) {
    const int bc = blockIdx.x, b = bc >> 2, c = bc & 3;
    __shared__ unsigned mbits[HH * 8];  // 1 bit per pixel of the slice (8KB)
    const int t = threadIdx.x;          // 256 threads; t = column in phase A, row in phase B
    const int lane = t & 31, w = t >> 5;

    // Phase A: coalesced load of y slice, pack mask bits via wave ballot.
    for (int r = 0; r < HH; ++r) {
        int val = y[b * HW + r * WW + t];
        unsigned long long bal = __ballot(val == c);
        if (lane == 0) mbits[r * 8 + w] = (unsigned)bal;
    }
    __syncthreads();

    // Phase B: thread t scans row t. All threads iterate jj in lockstep, so the
    // gT[jj*WW + t] global accesses are coalesced across the block.
    float* grow = gT + (size_t)bc * HW;
    unsigned mb[8];
#pragma unroll
    for (int q = 0; q < 8; ++q) mb[q] = mbits[t * 8 + q];

    int last = -100000;  // sentinel: no zero seen yet
    for (int q = 0; q < 8; ++q) {
        unsigned word = mb[q];
#pragma unroll
        for (int bpos = 0; bpos < 32; ++bpos) {
            int jj = q * 32 + bpos;
            if (((word >> bpos) & 1u) == 0u) last = jj;
            float d = (float)(jj - last);
            grow[jj * WW + t] = fminf(d * d, HBIG);
        }
    }
    int nxt = 100000;
    for (int q = 7; q >= 0; --q) {
        unsigned word = mb[q];
#pragma unroll
        for (int bpos = 31; bpos >= 0; --bpos) {
            int jj = q * 32 + bpos;
            if (((word >> bpos) & 1u) == 0u) nxt = jj;
            float d = (float)(nxt - jj);
            float cur = grow[jj * WW + t];
            grow[jj * WW + t] = fminf(cur, fminf(d * d, HBIG));
        }
    }
}

// ---------------------------------------------------------------------------
// Kernel 1b: column pass, one block per slice. dT[j][i] = min_k gT[j][k]+(i-k)^2
// In-place over gT (row j of dT depends only on row j of gT). Rows are staged
// into double-buffered LDS with async global->LDS copies overlapped with the
// 256-deep fma/min inner loop; async loads retire in order, so waiting to <=1
// outstanding guarantees the current row has landed.
// Also computes per-slice max of d^2 (for m = sqrt(max)+1e-15).
// ---------------------------------------------------------------------------
__global__ void edt_cols(float* __restrict__ gT, float* __restrict__ max_dsq) {
    const int bc = blockIdx.x;
    __shared__ float sh[2][WW];
    __shared__ float red[256];
    const int i = threadIdx.x;
    float* base = gT + (size_t)bc * HW;
    float mx = 0.f;

    async_load_f32_to_lds(&base[i], &sh[0][i]);  // prefetch row 0 -> buf 0

    for (int j = 0; j < HH; ++j) {
        const int buf = j & 1;
        if (j + 1 < HH) {
            // Safe to overwrite buf^1: every wave passed the post-compute
            // barrier of iteration j-1, which ended all reads of that buffer.
            async_load_f32_to_lds(&base[(j + 1) * WW + i], &sh[buf ^ 1][i]);
            asm volatile("s_wait_asynccnt 0x1" ::: "memory");  // row j has landed
        } else {
            asm volatile("s_wait_asynccnt 0x0" ::: "memory");
        }
        __syncthreads();  // row j visible to all waves

        const float fi = (float)i;
        float best = 2.0f * HBIG;
#pragma unroll 4
        for (int k = 0; k < WW; ++k) {
            float dk = fi - (float)k;
            best = fminf(best, fmaf(dk, dk, sh[buf][k]));
        }
        __syncthreads();  // all waves done reading sh[buf] before it is reused

        base[j * WW + i] = best;  // overwrite: gT row j becomes dT row j
        mx = fmaxf(mx, best);
    }

    red[i] = mx;
    __syncthreads();
    for (int s = 128; s > 0; s >>= 1) {
        if (i < s) red[i] = fmaxf(red[i], red[i + s]);
        __syncthreads();
    }
    if (i == 0) max_dsq[bc] = red[0];
}

// ---------------------------------------------------------------------------
// Kernel 2: per-pixel softmax, CE log term and sum_c p*rrwmap, reduced with
// V_WMMA_F32_16X16X4_F32 (A = 64 partials per wave-step, B = ones, C += ).
// ---------------------------------------------------------------------------
__device__ __forceinline__ void pixel_terms(const float* __restrict__ x,
                                            const int* __restrict__ y,
                                            const float* __restrict__ dT,
                                            const float* __restrict__ mlds,
                                            int p, float& celog, float& rw) {
    const int b = p >> 16, rem = p & (HW - 1);
    const int i = rem >> 8, j = rem & 255;
    const float* xb = x + (size_t)b * CC * HW + rem;
    float x0 = xb[0], x1 = xb[HW], x2 = xb[2 * HW], x3 = xb[3 * HW];
    float mxv = fmaxf(fmaxf(x0, x1), fmaxf(x2, x3));
    float e0 = expf(x0 - mxv), e1 = expf(x1 - mxv), e2 = expf(x2 - mxv), e3 = expf(x3 - mxv);
    float inv = 1.f / (e0 + e1 + e2 + e3);
    float pc[4] = {e0 * inv, e1 * inv, e2 * inv, e3 * inv};
    int lab = y[p];
    float py = (lab == 0) ? pc[0] : (lab == 1) ? pc[1] : (lab == 2) ? pc[2] : pc[3];
    celog = logf(py + 1e-15f);
    float acc = 0.f;
#pragma unroll
    for (int c = 0; c < 4; ++c) {
        int bc = b * 4 + c;
        float dsq = dT[(size_t)bc * HW + j * WW + i];
        float rr = (dsq == 0.f) ? 1.f : -(sqrtf(dsq) / mlds[bc]);
        acc = fmaf(pc[c], rr, acc);
    }
    rw = acc;
}

__global__ void loss_partials(const float* __restrict__ x, const int* __restrict__ y,
                              const float* __restrict__ dT, const float* __restrict__ max_dsq,
                              float* __restrict__ cePart, float* __restrict__ rwPart) {
    __shared__ float mlds[NSLICE];
    __shared__ float wce[8], wrw[8];
    const int t = threadIdx.x;  // 256 threads, 8 waves
    if (t < NSLICE) mlds[t] = sqrtf(max_dsq[t]) + 1e-15f;
    __syncthreads();

    const int lane = t & 31, wid = t >> 5;
    const int gw = (blockIdx.x * 256 + t) >> 5;  // global wave id, 512 total

    v8f cce = {0.f, 0.f, 0.f, 0.f, 0.f, 0.f, 0.f, 0.f};
    v8f crw = {0.f, 0.f, 0.f, 0.f, 0.f, 0.f, 0.f, 0.f};
    const v2f ones = {1.f, 1.f};

    // 512 waves x 8 iters x 64 pixels == 262144 pixels, each exactly once.
    for (int it = 0; it < 8; ++it) {
        int basep = (it * 512 + gw) * 64;
        float c0, r0, c1, r1;
        pixel_terms(x, y, dT, mlds, basep + lane, c0, r0);
        pixel_terms(x, y, dT, mlds, basep + 32 + lane, c1, r1);
        v2f ace = {c0, c1};
        v2f arw = {r0, r1};
        // D = A x ones + C : matrix-unit reduction of 64 f32 partials per op.
        cce = __builtin_amdgcn_wmma_f32_16x16x4_f32(false, ace, false, ones,
                                                    (short)0, cce, false, false);
        crw = __builtin_amdgcn_wmma_f32_16x16x4_f32(false, arw, false, ones,
                                                    (short)0, crw, false, false);
    }

    // D[i][j] = rowsum_i for every column j. Column 0 lives in lane 0 (M=0..7)
    // and lane 16 (M=8..15): total = sum of their 8 accumulator components.
    float sce = 0.f, srw = 0.f;
#pragma unroll
    for (int r = 0; r < 8; ++r) { sce += cce[r]; srw += crw[r]; }
    float tce = __shfl(sce, 0, 32) + __shfl(sce, 16, 32);
    float trw = __shfl(srw, 0, 32) + __shfl(srw, 16, 32);
    if (lane == 0) { wce[wid] = tce; wrw[wid] = trw; }
    __syncthreads();
    if (t == 0) {
        float a = 0.f, bsum = 0.f;
#pragma unroll
        for (int q = 0; q < 8; ++q) { a += wce[q]; bsum += wrw[q]; }
        cePart[blockIdx.x] = a;
        rwPart[blockIdx.x] = bsum;
    }
}

// ---------------------------------------------------------------------------
// Kernel 3: deterministic fixed-order final combine.
// ---------------------------------------------------------------------------
__global__ void finalize(const float* __restrict__ cePart, const float* __restrict__ rwPart,
                         float* __restrict__ out) {
    if (threadIdx.x == 0 && blockIdx.x == 0) {
        float sc = 0.f, sr = 0.f;
        for (int q = 0; q < 64; ++q) { sc += cePart[q]; sr += rwPart[q]; }
        out[0] = -sc / (float)NPIX + sr / (float)NELEM;
    }
}

extern "C" void kernel_launch(void* const* d_in, const int* in_sizes, int n_in,
                              void* d_out, int out_size, void* d_ws, size_t ws_size,
                              hipStream_t stream) {
    (void)in_sizes; (void)n_in; (void)out_size; (void)ws_size;
    const float* x = (const float*)d_in[0];   // [4,4,256,256] f32
    const int* y = (const int*)d_in[1];       // [4,1,256,256] i32
    // d_in[2] (e) is unused by the reference's math.

    float* wsf = (float*)d_ws;
    float* gT = wsf;                                // 16*65536 f32 (4MB); becomes dT in-place
    float* max_dsq = wsf + (size_t)NSLICE * HW;     // 16 f32
    float* cePart = max_dsq + NSLICE;               // 64 f32
    float* rwPart = cePart + 64;                    // 64 f32
    float* out = (float*)d_out;

    edt_rows<<<NSLICE, 256, 0, stream>>>(y, gT);
    edt_cols<<<NSLICE, 256, 0, stream>>>(gT, max_dsq);
    loss_partials<<<64, 256, 0, stream>>>(x, y, gT, max_dsq, cePart, rwPart);
    finalize<<<1, 32, 0, stream>>>(cePart, rwPart, out);
}